// IntLoss_69346541961703
// MI455X (gfx1250) — compile-verified
//
#include <hip/hip_runtime.h>
#include <hip/hip_bf16.h>
#include <math.h>

#define BLOCK 256

typedef __attribute__((ext_vector_type(4))) float v4f;
typedef __attribute__((ext_vector_type(4))) int v4i;
// Probe-confirmed builtin signature: (int4 AS1*, int4 AS3*, imm off, imm cpol).
typedef __attribute__((address_space(1))) v4i gas_v4i;
typedef __attribute__((address_space(3))) v4i las_v4i;

#if defined(__gfx1250__) && __has_builtin(__builtin_amdgcn_global_load_async_to_lds_b128)
#define HAVE_ASYNC_LDS 1
#else
#define HAVE_ASYNC_LDS 0
#endif

__device__ __forceinline__ void async_cp16(const v4f* g, v4f* l) {
#if HAVE_ASYNC_LDS
  __builtin_amdgcn_global_load_async_to_lds_b128((gas_v4i*)g, (las_v4i*)l, 0, 0);
#endif
}

// Wait until at most n async-to-LDS ops are outstanding (ASYNCcnt <= n).
// n is a compile-time constant at every call site (unrolled loop), so the
// if-chain folds to a single s_wait_asynccnt with an immediate.
__device__ __forceinline__ void async_wait(int n) {
#if __has_builtin(__builtin_amdgcn_s_wait_asynccnt)
  if (n <= 0)
    __builtin_amdgcn_s_wait_asynccnt(0);
  else if (n == 1)
    __builtin_amdgcn_s_wait_asynccnt(1);
  else if (n == 2)
    __builtin_amdgcn_s_wait_asynccnt(2);
  else
    __builtin_amdgcn_s_wait_asynccnt(3);
#elif defined(__gfx1250__)
  if (n <= 0)
    asm volatile("s_wait_asynccnt 0x0" ::: "memory");
  else if (n == 1)
    asm volatile("s_wait_asynccnt 0x1" ::: "memory");
  else if (n == 2)
    asm volatile("s_wait_asynccnt 0x2" ::: "memory");
  else
    asm volatile("s_wait_asynccnt 0x3" ::: "memory");
#endif
}

// Chunk-level online softmax: one max over the float4, a rare rescale branch
// (running max only grows a few times per row), then 4 independent exps.
__device__ __forceinline__ void updc(float& m, float& s, v4f x) {
  float cm = fmaxf(fmaxf(x.x, x.y), fmaxf(x.z, x.w));
  if (cm > m) {
    s *= __expf(m - cm);  // m == -FLT_MAX initially -> exp(-inf) == 0
    m = cm;
  }
  s += __expf(x.x - m) + __expf(x.y - m) + __expf(x.z - m) + __expf(x.w - m);
}
__device__ __forceinline__ void upd1(float& m, float& s, float x) {
  if (x > m) {
    s = s * __expf(m - x) + 1.0f;
    m = x;
  } else {
    s += __expf(x - m);
  }
}

// One workgroup per row: online logsumexp over CC f32 logits streamed through
// a 4-deep async-to-LDS pipeline, then gather L labels, write per-row partial.
template <int CC>
__global__ __launch_bounds__(BLOCK) void ce_row_kernel(
    const float* __restrict__ logits, const long long* __restrict__ labels,
    float* __restrict__ partial, int L) {
  constexpr int NV = CC / 4;       // float4 per row (2500)
  constexpr int NF = NV / BLOCK;   // full tiles (9)
  constexpr int NR = NV % BLOCK;   // ragged tail in float4 (196)
  constexpr int D = (NF >= 4) ? 4 : (NF > 0 ? NF : 1);  // pipeline depth

  const int b = blockIdx.x;
  const int tid = threadIdx.x;
  const float* __restrict__ row = logits + (size_t)b * (size_t)CC;
  const v4f* __restrict__ rowv = (const v4f*)row;

  __shared__ float rm[BLOCK];
  __shared__ float rs[BLOCK];
  __shared__ float shZ;
  __shared__ float gval[32];
  __shared__ int glab[32];

  float m = -3.402823466e+38f;  // -FLT_MAX
  float s = 0.0f;

#if HAVE_ASYNC_LDS
  __shared__ v4f buf[D][BLOCK];
  // Prologue: put D-1 tiles in flight. Slot rotation keeps a one-iteration
  // gap between the DS read of a slot and its async rewrite (WAR-safe: the
  // consumed value is already in registers, DScnt-waited, before reissue).
#pragma unroll
  for (int p = 0; p < D - 1; ++p) {
    if (p < NF) async_cp16(rowv + p * BLOCK + tid, &buf[p][tid]);
  }
#pragma unroll
  for (int t = 0; t < NF; ++t) {
    if (t + D - 1 < NF)
      async_cp16(rowv + (t + D - 1) * BLOCK + tid, &buf[(t + D - 1) % D][tid]);
    // outstanding = min(NF, t+D) - t; tile t landed once ASYNCcnt <= that-1
    const int out_after = ((t + D) < NF ? (t + D) : NF) - t;
    async_wait(out_after - 1);
    v4f x = buf[t % D][tid];
    updc(m, s, x);
  }
#else
#pragma unroll
  for (int t = 0; t < NF; ++t) {
    v4f x = rowv[t * BLOCK + tid];
    updc(m, s, x);
  }
#endif
  // Ragged tail (2500 = 9*256 + 196): direct b128 loads, masked lanes.
  if (NR > 0 && tid < NR) {
    v4f x = rowv[NF * BLOCK + tid];
    updc(m, s, x);
  }
  // Scalar tail if CC % 4 != 0 (none for CC=10000).
  for (int c = (NV << 2) + tid; c < CC; c += BLOCK) upd1(m, s, row[c]);

  // Combine 256 (m, s) pairs with the online-softmax merge rule.
  rm[tid] = m;
  rs[tid] = s;
  __syncthreads();
#pragma unroll
  for (int off = BLOCK / 2; off > 0; off >>= 1) {
    if (tid < off) {
      float m1 = rm[tid], s1 = rs[tid];
      float m2 = rm[tid + off], s2 = rs[tid + off];
      float M = fmaxf(m1, m2);
      rs[tid] = s1 * __expf(m1 - M) + s2 * __expf(m2 - M);
      rm[tid] = M;
    }
    __syncthreads();
  }
  if (tid == 0) shZ = rm[0] + logf(rs[0]);

  // Parallel gather of the L label logits; prefix-validity (break at first
  // label < 0) resolved serially by lane 0 (L == 20, negligible).
  if (tid < L && tid < 32) {
    long long lab = labels[(long long)b * L + tid];
    glab[tid] = (int)lab;
    gval[tid] = (lab >= 0 && lab < (long long)CC) ? row[(int)lab] : 0.0f;
  }
  __syncthreads();
  if (tid == 0) {
    float logZ = shZ;
    float acc = 0.0f;
    for (int l = 0; l < L; ++l) {
      int la = glab[l];
      if (la < 0) break;  // reference: cumprod(labels >= 0) prefix mask
      acc += logZ - gval[l];
    }
    partial[b] = acc;  // contribution to  -sum(gathered * valid)
  }
}

// Deterministic final reduction: fixed per-thread serial order + fixed tree.
__global__ __launch_bounds__(BLOCK) void final_reduce_kernel(
    const float* __restrict__ partial, float* __restrict__ out, int B) {
  __shared__ float r[BLOCK];
  float acc = 0.0f;
  for (int i = threadIdx.x; i < B; i += BLOCK) acc += partial[i];
  r[threadIdx.x] = acc;
  __syncthreads();
#pragma unroll
  for (int off = BLOCK / 2; off > 0; off >>= 1) {
    if (threadIdx.x < off) r[threadIdx.x] += r[threadIdx.x + off];
    __syncthreads();
  }
  if (threadIdx.x == 0) out[0] = r[0];
}

extern "C" void kernel_launch(void* const* d_in, const int* in_sizes, int n_in,
                              void* d_out, int out_size, void* d_ws,
                              size_t ws_size, hipStream_t stream) {
  const float* logits = (const float*)d_in[0];          // [B, C] f32
  const long long* labels = (const long long*)d_in[1];  // [B, L] i64
  float* out = (float*)d_out;                           // scalar f32

  const int C = 10000;
  const int B = in_sizes[0] / C;                        // 16384
  const int L = (B > 0) ? (in_sizes[1] / B) : 0;        // 20

  float* partial = (float*)d_ws;                        // B floats of scratch

  ce_row_kernel<10000><<<B, BLOCK, 0, stream>>>(logits, labels, partial, L);
  final_reduce_kernel<<<1, BLOCK, 0, stream>>>(partial, out, B);
}